// GATGraphClassifier_51488067944914
// MI455X (gfx1250) — compile-verified
//
#include <hip/hip_runtime.h>
#include <hip/hip_bf16.h>

#define N_NODES 50000
#define N_EDGES 800000
#define E2      (N_EDGES + N_NODES)   // edges + self loops = 850000
#define N_GRAPHS 64
#define IN_DIM  128
#define HID     32
#define HEADS   4
#define EDGE_DIM 8
#define OUTC    10

typedef __attribute__((ext_vector_type(2))) float v2f;
typedef __attribute__((ext_vector_type(8))) float v8f;

static __device__ __forceinline__ unsigned fkey(float f) {
    unsigned u = __float_as_uint(f);
    return (u & 0x80000000u) ? ~u : (u | 0x80000000u);
}
static __device__ __forceinline__ float funkey(unsigned k) {
    unsigned u = (k & 0x80000000u) ? (k & 0x7FFFFFFFu) : ~k;
    return __uint_as_float(u);
}
static __device__ __forceinline__ float leaky02(float a) { return a > 0.f ? a : 0.2f * a; }
static __device__ __forceinline__ float elu1(float a)    { return a > 0.f ? a : (__expf(a) - 1.f); }

// ---------------------------------------------------------------------------
// 1) degree + sum of incoming edge_attr per dst (for self-loop fill 'mean')
// ---------------------------------------------------------------------------
__global__ void edge_degree_kernel(const int* __restrict__ ei,
                                   const float* __restrict__ eattr,
                                   float* __restrict__ cnt,
                                   float* __restrict__ loopattr) {
    int e = blockIdx.x * blockDim.x + threadIdx.x;
    if (e >= N_EDGES) return;
    int d = ei[N_EDGES + e];
    atomicAdd(&cnt[d], 1.0f);
    const float* a = eattr + (size_t)e * EDGE_DIM;
    #pragma unroll
    for (int k = 0; k < EDGE_DIM; ++k)
        atomicAdd(&loopattr[(size_t)d * EDGE_DIM + k], a[k]);
}

__global__ void loop_finalize_kernel(const float* __restrict__ cnt,
                                     float* __restrict__ loopattr) {
    int t = blockIdx.x * blockDim.x + threadIdx.x;
    if (t >= N_NODES * EDGE_DIM) return;
    float c = cnt[t >> 3];
    loopattr[t] /= (c > 1.f ? c : 1.f);
}

// ---------------------------------------------------------------------------
// 2) fold We@att_e into tiny vectors: v1[8][4], v2[8]
// ---------------------------------------------------------------------------
__global__ void fold_edge_att_kernel(const float* __restrict__ We1, const float* __restrict__ ae1,
                                     const float* __restrict__ We2, const float* __restrict__ ae2,
                                     float* __restrict__ v1, float* __restrict__ v2) {
    int t = threadIdx.x;
    if (t < EDGE_DIM * HEADS) {            // v1[d*4+h]
        int d = t >> 2, h = t & 3;
        float acc = 0.f;
        for (int c = 0; c < HID; ++c)
            acc += We1[d * (HEADS * HID) + h * HID + c] * ae1[h * HID + c];
        v1[t] = acc;
    } else if (t < EDGE_DIM * HEADS + EDGE_DIM) {  // v2[d]
        int d = t - EDGE_DIM * HEADS;
        float acc = 0.f;
        for (int c = 0; c < HID; ++c)
            acc += We2[d * HID + c] * ae2[c];
        v2[d] = acc;
    }
}

// ---------------------------------------------------------------------------
// 3) fp32 WMMA GEMM: C[M x Ncols] = A[M x K] * B[K x Ncols], row-major.
//    One wave per 16x16 tile, V_WMMA_F32_16X16X4_F32, K stepped by 4.
//    A frag (16x4): lane half selects K+2; B frag (4x16): row K = vgpr + 2*half.
// ---------------------------------------------------------------------------
__global__ __launch_bounds__(32)
void wmma_gemm_f32_kernel(const float* __restrict__ A, const float* __restrict__ B,
                          float* __restrict__ C, int K, int Ncols) {
    const int row0 = blockIdx.x * 16;
    const int col0 = blockIdx.y * 16;
    const int l    = threadIdx.x;
    const int half = l >> 4;
    const int lm   = l & 15;
    const float* arow = A + (size_t)(row0 + lm) * K + 2 * half;
    const float* bptr = B + (size_t)(2 * half) * Ncols + col0 + lm;
    v8f c = {0.f, 0.f, 0.f, 0.f, 0.f, 0.f, 0.f, 0.f};
    for (int k = 0; k < K; k += 4) {
        v2f a, b;
        a.x = arow[k];
        a.y = arow[k + 1];
        b.x = bptr[(size_t)k * Ncols];
        b.y = bptr[(size_t)(k + 1) * Ncols];
        c = __builtin_amdgcn_wmma_f32_16x16x4_f32(false, a, false, b, (short)0, c,
                                                  false, false);
    }
    #pragma unroll
    for (int i = 0; i < 8; ++i)
        C[(size_t)(row0 + i + 8 * half) * Ncols + col0 + lm] = c[i];
}

// ---------------------------------------------------------------------------
// 4) per-node attention logits: als[n,h] = h[n,h,:]·att_src, ald likewise
// ---------------------------------------------------------------------------
__global__ void node_att_kernel(const float* __restrict__ h,
                                const float* __restrict__ att_s,
                                const float* __restrict__ att_d,
                                float* __restrict__ als, float* __restrict__ ald,
                                int H) {
    int t = blockIdx.x * blockDim.x + threadIdx.x;
    if (t >= N_NODES * H) return;
    int n = t / H, hh = t % H;
    const float* row = h + (size_t)n * H * HID + hh * HID;
    float a = 0.f, b = 0.f;
    #pragma unroll
    for (int c = 0; c < HID; ++c) {
        float x = row[c];
        a += x * att_s[hh * HID + c];
        b += x * att_d[hh * HID + c];
    }
    als[t] = a;
    ald[t] = b;
}

// ---------------------------------------------------------------------------
// 5) edge passes (shared for both convs via H template-ish param)
// ---------------------------------------------------------------------------
static __device__ __forceinline__ void edge_fetch(int e, const int* ei,
                                                  const float* eattr, const float* loopattr,
                                                  int& s, int& d, const float*& attr) {
    if (e < N_EDGES) {
        s = ei[e];
        d = ei[N_EDGES + e];
        attr = eattr + (size_t)e * EDGE_DIM;
    } else {
        s = d = e - N_EDGES;
        attr = loopattr + (size_t)(e - N_EDGES) * EDGE_DIM;
    }
}

__global__ void edge_alpha_kernel(const int* __restrict__ ei,
                                  const float* __restrict__ eattr,
                                  const float* __restrict__ loopattr,
                                  const float* __restrict__ als,
                                  const float* __restrict__ ald,
                                  const float* __restrict__ vfold,   // [EDGE_DIM * H]
                                  float* __restrict__ alpha,         // [E2 * H]
                                  unsigned* __restrict__ mkey,       // [N * H]
                                  int H) {
    int e = blockIdx.x * blockDim.x + threadIdx.x;
    if (e >= E2) return;
    int s, d; const float* attr;
    edge_fetch(e, ei, eattr, loopattr, s, d, attr);
    float at[EDGE_DIM];
    #pragma unroll
    for (int k = 0; k < EDGE_DIM; ++k) at[k] = attr[k];
    for (int hh = 0; hh < H; ++hh) {
        float ae = 0.f;
        #pragma unroll
        for (int k = 0; k < EDGE_DIM; ++k) ae += at[k] * vfold[k * H + hh];
        float a = leaky02(als[s * H + hh] + ald[d * H + hh] + ae);
        alpha[(size_t)e * H + hh] = a;
        atomicMax(&mkey[d * H + hh], fkey(a));
    }
}

__global__ void edge_exp_kernel(const int* __restrict__ ei,
                                float* __restrict__ alpha,
                                const unsigned* __restrict__ mkey,
                                float* __restrict__ denom, int H) {
    int e = blockIdx.x * blockDim.x + threadIdx.x;
    if (e >= E2) return;
    int d = (e < N_EDGES) ? ei[N_EDGES + e] : (e - N_EDGES);
    for (int hh = 0; hh < H; ++hh) {
        float m = funkey(mkey[d * H + hh]);
        float w = __expf(alpha[(size_t)e * H + hh] - m);
        alpha[(size_t)e * H + hh] = w;
        atomicAdd(&denom[d * H + hh], w);
    }
}

// one thread per (edge, output channel); HC = H*HID total channels
__global__ void edge_msg_kernel(const int* __restrict__ ei,
                                const float* __restrict__ hsrc,   // [N, HC]
                                const float* __restrict__ alpha,  // [E2, H]
                                const float* __restrict__ denom,  // [N, H]
                                float* __restrict__ out,          // [N, HC]
                                int H, int HC) {
    long long t = (long long)blockIdx.x * blockDim.x + threadIdx.x;
    if (t >= (long long)E2 * HC) return;
    int e = (int)(t / HC);
    int j = (int)(t % HC);
    int hh = j / HID;
    int s, d;
    if (e < N_EDGES) { s = ei[e]; d = ei[N_EDGES + e]; }
    else             { s = d = e - N_EDGES; }
    float w = alpha[(size_t)e * H + hh] / (denom[d * H + hh] + 1e-16f);
    atomicAdd(&out[(size_t)d * HC + j], hsrc[(size_t)s * HC + j] * w);
}

__global__ void bias_elu_kernel(float* __restrict__ x, const float* __restrict__ b, int HC) {
    long long t = (long long)blockIdx.x * blockDim.x + threadIdx.x;
    if (t >= (long long)N_NODES * HC) return;
    float v = x[t] + b[(int)(t % HC)];
    x[t] = elu1(v);
}

__global__ void bias_elu_pool_kernel(const float* __restrict__ x, const float* __restrict__ b,
                                     const int* __restrict__ batch,
                                     float* __restrict__ gsum, float* __restrict__ gcnt) {
    int t = blockIdx.x * blockDim.x + threadIdx.x;
    if (t >= N_NODES * HID) return;
    int n = t >> 5, j = t & 31;
    float v = elu1(x[t] + b[j]);
    int g = batch[n];
    atomicAdd(&gsum[g * HID + j], v);
    if (j == 0) atomicAdd(&gcnt[g], 1.0f);
}

// ---------------------------------------------------------------------------
// 6) tiny MLP head, one thread per graph
// ---------------------------------------------------------------------------
__global__ void mlp_head_kernel(const float* __restrict__ gsum, const float* __restrict__ gcnt,
                                const float* __restrict__ W3, const float* __restrict__ b3,
                                const float* __restrict__ W4, const float* __restrict__ b4,
                                float* __restrict__ out) {
    int g = threadIdx.x;
    if (g >= N_GRAPHS) return;
    float c = gcnt[g]; c = c > 1.f ? c : 1.f;
    float gv[HID];
    #pragma unroll
    for (int j = 0; j < HID; ++j) gv[j] = gsum[g * HID + j] / c;
    float z[HID / 2];
    #pragma unroll
    for (int j = 0; j < HID / 2; ++j) {
        float acc = b3[j];
        for (int k = 0; k < HID; ++k) acc += gv[k] * W3[k * (HID / 2) + j];
        z[j] = acc > 0.f ? acc : 0.f;
    }
    for (int o = 0; o < OUTC; ++o) {
        float acc = b4[o];
        #pragma unroll
        for (int j = 0; j < HID / 2; ++j) acc += z[j] * W4[j * OUTC + o];
        out[g * OUTC + o] = acc;
    }
}

static inline int cdiv(long long a, long long b) { return (int)((a + b - 1) / b); }

extern "C" void kernel_launch(void* const* d_in, const int* in_sizes, int n_in,
                              void* d_out, int out_size, void* d_ws, size_t ws_size,
                              hipStream_t stream) {
    const float* x      = (const float*)d_in[0];
    const int*   ei     = (const int*)  d_in[1];
    const float* eattr  = (const float*)d_in[2];
    const int*   batch  = (const int*)  d_in[3];
    const float* W1     = (const float*)d_in[4];
    const float* as1    = (const float*)d_in[5];
    const float* ad1    = (const float*)d_in[6];
    const float* We1    = (const float*)d_in[7];
    const float* ae1    = (const float*)d_in[8];
    const float* b1     = (const float*)d_in[9];
    const float* W2     = (const float*)d_in[10];
    const float* as2    = (const float*)d_in[11];
    const float* ad2    = (const float*)d_in[12];
    const float* We2    = (const float*)d_in[13];
    const float* ae2    = (const float*)d_in[14];
    const float* b2     = (const float*)d_in[15];
    const float* W3     = (const float*)d_in[16];
    const float* b3     = (const float*)d_in[17];
    const float* W4     = (const float*)d_in[18];
    const float* b4     = (const float*)d_in[19];
    float* out = (float*)d_out;

    // ---- workspace layout (floats), buffers reused between conv phases ----
    float* ws = (float*)d_ws;
    float*    cnt      = ws;                               // N
    float*    loopattr = cnt + N_NODES;                    // N*8
    float*    h1       = loopattr + (size_t)N_NODES * 8;   // N*128 (reused: h2 in first N*32)
    float*    out1     = h1 + (size_t)N_NODES * 128;       // N*128 (reused: out2 in first N*32)
    float*    als      = out1 + (size_t)N_NODES * 128;     // N*4 (conv2 uses first N)
    float*    ald      = als + (size_t)N_NODES * 4;        // N*4
    float*    alpha    = ald + (size_t)N_NODES * 4;        // E2*4 (conv2 uses first E2)
    unsigned* mkey     = (unsigned*)(alpha + (size_t)E2 * 4); // N*4
    float*    denom    = (float*)(mkey + (size_t)N_NODES * 4); // N*4
    float*    v1       = denom + (size_t)N_NODES * 4;      // 32
    float*    v2       = v1 + 32;                          // 8
    float*    gsum     = v2 + 8;                           // 64*32
    float*    gcnt     = gsum + N_GRAPHS * HID;            // 64

    const int B = 256;

    // ---- self-loop attrs (fill_value='mean') ----
    hipMemsetAsync(cnt, 0, sizeof(float) * N_NODES, stream);
    hipMemsetAsync(loopattr, 0, sizeof(float) * (size_t)N_NODES * 8, stream);
    edge_degree_kernel<<<cdiv(N_EDGES, B), B, 0, stream>>>(ei, eattr, cnt, loopattr);
    loop_finalize_kernel<<<cdiv((long long)N_NODES * 8, B), B, 0, stream>>>(cnt, loopattr);

    // ---- fold edge attention weights ----
    fold_edge_att_kernel<<<1, 64, 0, stream>>>(We1, ae1, We2, ae2, v1, v2);

    // ================= conv1 (H=4, HC=128) =================
    // h1 = x @ W1   [50000x128] @ [128x128]
    {
        dim3 grid(N_NODES / 16, (HEADS * HID) / 16);
        wmma_gemm_f32_kernel<<<grid, 32, 0, stream>>>(x, W1, h1, IN_DIM, HEADS * HID);
    }
    node_att_kernel<<<cdiv((long long)N_NODES * HEADS, B), B, 0, stream>>>(h1, as1, ad1, als, ald, HEADS);
    hipMemsetAsync(mkey, 0, sizeof(unsigned) * (size_t)N_NODES * HEADS, stream);
    edge_alpha_kernel<<<cdiv(E2, B), B, 0, stream>>>(ei, eattr, loopattr, als, ald, v1, alpha, mkey, HEADS);
    hipMemsetAsync(denom, 0, sizeof(float) * (size_t)N_NODES * HEADS, stream);
    edge_exp_kernel<<<cdiv(E2, B), B, 0, stream>>>(ei, alpha, mkey, denom, HEADS);
    hipMemsetAsync(out1, 0, sizeof(float) * (size_t)N_NODES * 128, stream);
    edge_msg_kernel<<<cdiv((long long)E2 * 128, B), B, 0, stream>>>(ei, h1, alpha, denom, out1, HEADS, 128);
    bias_elu_kernel<<<cdiv((long long)N_NODES * 128, B), B, 0, stream>>>(out1, b1, 128);

    // ================= conv2 (H=1, HC=32) =================
    float* h2   = h1;    // reuse, N*32
    float* out2 = out1 + (size_t)N_NODES * 128 - (size_t)N_NODES * 128; // alias of out1
    // h2 = elu(out1) @ W2   [50000x128] @ [128x32]
    {
        dim3 grid(N_NODES / 16, HID / 16);
        wmma_gemm_f32_kernel<<<grid, 32, 0, stream>>>(out1, W2, h2, HEADS * HID, HID);
    }
    node_att_kernel<<<cdiv(N_NODES, B), B, 0, stream>>>(h2, as2, ad2, als, ald, 1);
    hipMemsetAsync(mkey, 0, sizeof(unsigned) * (size_t)N_NODES, stream);
    edge_alpha_kernel<<<cdiv(E2, B), B, 0, stream>>>(ei, eattr, loopattr, als, ald, v2, alpha, mkey, 1);
    hipMemsetAsync(denom, 0, sizeof(float) * (size_t)N_NODES, stream);
    edge_exp_kernel<<<cdiv(E2, B), B, 0, stream>>>(ei, alpha, mkey, denom, 1);
    hipMemsetAsync(out2, 0, sizeof(float) * (size_t)N_NODES * HID, stream);
    edge_msg_kernel<<<cdiv((long long)E2 * HID, B), B, 0, stream>>>(ei, h2, alpha, denom, out2, 1, HID);

    // ---- ELU + global mean pool + MLP head ----
    hipMemsetAsync(gsum, 0, sizeof(float) * N_GRAPHS * HID, stream);
    hipMemsetAsync(gcnt, 0, sizeof(float) * N_GRAPHS, stream);
    bias_elu_pool_kernel<<<cdiv((long long)N_NODES * HID, B), B, 0, stream>>>(out2, b2, batch, gsum, gcnt);
    mlp_head_kernel<<<1, 64, 0, stream>>>(gsum, gcnt, W3, b3, W4, b4, out);
}